// CrossAttention_62749472195291
// MI455X (gfx1250) — compile-verified
//
#include <hip/hip_runtime.h>
#include <math.h>

// ---------------------------------------------------------------------------
// Cross-attention, MI455X (gfx1250, wave32).
// Memory-bound (~192MB min traffic @ 23.3TB/s -> ~8us floor). Flash-style
// attention: the 128MB score matrix is never materialized. Scores computed
// with V_WMMA_F32_16X16X4_F32 (full fp32, matches reference precision).
// K/V staged into LDS with double-buffered GLOBAL_LOAD_ASYNC_TO_LDS
// (ASYNCcnt-tracked), overlapping global->LDS traffic with WMMA compute.
// ---------------------------------------------------------------------------

typedef __attribute__((ext_vector_type(2))) float v2f;
typedef __attribute__((ext_vector_type(8))) float v8f;

// Async-copy builtin operand types (match clang's declared signatures:
// b64 -> int __vector(2)*, b128 -> int __vector(4)*).
typedef int vi2 __attribute__((vector_size(8)));
typedef int vi4 __attribute__((vector_size(16)));

#define BATCH  8
#define SEQ    2048
#define DMODEL 1024
#define HD     8
#define ROWS   (BATCH * SEQ)   // 16384
#define CHUNK  256             // K/V rows staged in LDS per iteration
#define KSTR   10              // padded floats per K row in LDS (bank-conflict free)

#if __has_builtin(__builtin_amdgcn_global_load_async_to_lds_b64) && \
    __has_builtin(__builtin_amdgcn_global_load_async_to_lds_b128)
#define USE_ASYNC_LDS 1
#else
#define USE_ASYNC_LDS 0
#endif

#if USE_ASYNC_LDS
__device__ __forceinline__ void async_cp_b64(const void* g, void* l) {
    __builtin_amdgcn_global_load_async_to_lds_b64(
        (__attribute__((address_space(1))) vi2*)g,
        (__attribute__((address_space(3))) vi2*)l, 0, 0);
}
__device__ __forceinline__ void async_cp_b128(const void* g, void* l) {
    __builtin_amdgcn_global_load_async_to_lds_b128(
        (__attribute__((address_space(1))) vi4*)g,
        (__attribute__((address_space(3))) vi4*)l, 0, 0);
}
#endif

__device__ __forceinline__ void wait_async_lds() {
#if USE_ASYNC_LDS
#if __has_builtin(__builtin_amdgcn_s_wait_asynccnt)
    __builtin_amdgcn_s_wait_asynccnt(0);
#else
    asm volatile("s_wait_asynccnt 0x0" ::: "memory");
#endif
#endif
}

// ---------------------------------------------------------------------------
// Kernel 1: projections. out = in[R,1024] @ W[1024,8] + b. Optionally a second
// (W2,b2,out2) pair computed from the SAME read of `in` (fuses k and v so
// context is read from HBM only once).
// Lane decomposition: h = lane&7 (output column), rs = lane>>3 (row slice).
// Each lane owns complete dot products -> no reductions, no LDS.
// W loads are uniform-in-d, h-indexed -> 32B broadcast reads (L2 resident).
// ---------------------------------------------------------------------------
__global__ __launch_bounds__(128)
void proj_qkv(const float* __restrict__ in,
              const float* __restrict__ W1, const float* __restrict__ b1,
              float* __restrict__ o1,
              const float* __restrict__ W2, const float* __restrict__ b2,
              float* __restrict__ o2, int dual)
{
    const int wave = threadIdx.x >> 5;
    const int lane = threadIdx.x & 31;
    const int h  = lane & 7;
    const int rs = lane >> 3;                     // 0..3
    const int r0 = (blockIdx.x * 4 + wave) * 16;  // 16 rows per wave

    const float4* in4 = (const float4*)in;

    float acc1[4], acc2[4];
    const float bb1 = b1[h];
    const float bb2 = dual ? b2[h] : 0.0f;
#pragma unroll
    for (int k = 0; k < 4; ++k) { acc1[k] = bb1; acc2[k] = bb2; }

    for (int i = 0; i < DMODEL / 4; ++i) {
        const int d = i * 4;
        const float wa = W1[(d + 0) * HD + h];
        const float wb = W1[(d + 1) * HD + h];
        const float wc = W1[(d + 2) * HD + h];
        const float wd = W1[(d + 3) * HD + h];
        float va = 0.f, vb = 0.f, vc = 0.f, vd = 0.f;
        if (dual) {
            va = W2[(d + 0) * HD + h];
            vb = W2[(d + 1) * HD + h];
            vc = W2[(d + 2) * HD + h];
            vd = W2[(d + 3) * HD + h];
        }
#pragma unroll
        for (int k = 0; k < 4; ++k) {
            const size_t row = (size_t)(r0 + rs + 4 * k);
            const float4 xv = in4[row * (DMODEL / 4) + i];
            acc1[k] = fmaf(xv.x, wa, acc1[k]);
            acc1[k] = fmaf(xv.y, wb, acc1[k]);
            acc1[k] = fmaf(xv.z, wc, acc1[k]);
            acc1[k] = fmaf(xv.w, wd, acc1[k]);
            if (dual) {
                acc2[k] = fmaf(xv.x, va, acc2[k]);
                acc2[k] = fmaf(xv.y, vb, acc2[k]);
                acc2[k] = fmaf(xv.z, vc, acc2[k]);
                acc2[k] = fmaf(xv.w, vd, acc2[k]);
            }
        }
    }
#pragma unroll
    for (int k = 0; k < 4; ++k) {
        const size_t row = (size_t)(r0 + rs + 4 * k);
        o1[row * HD + h] = acc1[k];               // 128B coalesced per store
        if (dual) o2[row * HD + h] = acc2[k];
    }
}

// ---------------------------------------------------------------------------
// Kernel 2: flash attention. Block = 4 waves; each wave owns a 16-query-row
// tile of one batch. K/V double-buffered in LDS in 256-row chunks, filled by
// async global->LDS copies (ASYNCcnt) overlapped with compute.
// Per 16-key tile: S^T = K_tile(16x8) x Q^T(8x16) via two chained
// v_wmma_f32_16x16x4_f32 (K-dim = h, split 0..3 / 4..7).
// C/D layout: lane L holds S^T[t0 + j + 8*(L>>4), s0 + (L&15)] in c[j].
// => each query row lives in one lane + its xor-16 partner: online softmax
// needs only per-lane reductions + one __shfl_xor(16).
// ---------------------------------------------------------------------------
__global__ __launch_bounds__(128)
void attn_kernel(const float* __restrict__ q, const float* __restrict__ k,
                 const float* __restrict__ v, float* __restrict__ attn)
{
    __shared__ float Ks[2][CHUNK * KSTR];   // 2 x 10 KB, padded rows
    __shared__ float Vs[2][CHUNK * HD];     // 2 x  8 KB

    const int b    = blockIdx.y;
    const int wave = threadIdx.x >> 5;
    const int lane = threadIdx.x & 31;
    const int n    = lane & 15;          // N index (query col of S^T)
    const int half = lane >> 4;          // lane half selects K/M sub-range
    const int s0   = blockIdx.x * 64 + wave * 16;

    const float* qb = q + (size_t)b * SEQ * HD;
    const float* kb = k + (size_t)b * SEQ * HD;
    const float* vb = v + (size_t)b * SEQ * HD;

    // B operand (Q^T, 8x16), loaded once per wave.
    // B layout (4x16 per wmma): VGPR0: K = 2*half, VGPR1: K = 2*half+1, N = lane&15.
    const float* qrow = qb + (size_t)(s0 + n) * HD;
    const v2f bq0 = *(const v2f*)(qrow + 2 * half);       // h = {2h0, 2h0+1}
    const v2f bq1 = *(const v2f*)(qrow + 4 + 2 * half);   // h = {4+2h0, 5+2h0}

    float m_state = -1e30f;
    float l_state = 0.0f;
    float acc[HD];
#pragma unroll
    for (int hh = 0; hh < HD; ++hh) acc[hh] = 0.0f;

    // ---- staging helper: copy CHUNK rows of K (padded) and V into buffer ----
    auto stage = [&](int tc, int buf) {
        for (int r = threadIdx.x; r < CHUNK; r += 128) {
            const float* krow = kb + (size_t)(tc + r) * HD;
            float*       kd   = &Ks[buf][r * KSTR];
            const float* vrow = vb + (size_t)(tc + r) * HD;
            float*       vd   = &Vs[buf][r * HD];
#if USE_ASYNC_LDS
            async_cp_b64(krow + 0, kd + 0);     // 8B-aligned (stride 40B rows)
            async_cp_b64(krow + 2, kd + 2);
            async_cp_b64(krow + 4, kd + 4);
            async_cp_b64(krow + 6, kd + 6);
            async_cp_b128(vrow + 0, vd + 0);    // 16B-aligned (stride 32B rows)
            async_cp_b128(vrow + 4, vd + 4);
#else
            const float4 ka = ((const float4*)krow)[0];
            const float4 kc = ((const float4*)krow)[1];
            *(v2f*)(kd + 0) = (v2f){ka.x, ka.y};
            *(v2f*)(kd + 2) = (v2f){ka.z, ka.w};
            *(v2f*)(kd + 4) = (v2f){kc.x, kc.y};
            *(v2f*)(kd + 6) = (v2f){kc.z, kc.w};
            ((float4*)vd)[0] = ((const float4*)vrow)[0];
            ((float4*)vd)[1] = ((const float4*)vrow)[1];
#endif
        }
    };

    // Preload chunk 0 into buffer 0.
    stage(0, 0);
    wait_async_lds();
    __syncthreads();

    int cur = 0;
    for (int tc = 0; tc < SEQ; tc += CHUNK) {
        // Kick off next chunk into the alternate buffer (overlaps compute).
        if (tc + CHUNK < SEQ) stage(tc + CHUNK, cur ^ 1);

        for (int t0 = 0; t0 < CHUNK; t0 += 16) {
            // A operand (K tile, 16x4 per wmma): lane L -> row t0 + (L&15),
            // VGPR0: K = 2*half, VGPR1: K = 2*half+1.
            const float* krow0 = &Ks[cur][(t0 + n) * KSTR];
            const v2f a0 = *(const v2f*)(krow0 + 2 * half);
            const v2f a1 = *(const v2f*)(krow0 + 4 + 2 * half);

            v8f c = {};
            c = __builtin_amdgcn_wmma_f32_16x16x4_f32(false, a0, false, bq0,
                                                      (short)0, c, false, false);
            c = __builtin_amdgcn_wmma_f32_16x16x4_f32(false, a1, false, bq1,
                                                      (short)0, c, false, false);
            // c[j] = score(s = s0+n, t = tc + t0 + j + 8*half)

            float tmax = c[0];
#pragma unroll
            for (int j = 1; j < 8; ++j) tmax = fmaxf(tmax, c[j]);
            tmax = fmaxf(tmax, __shfl_xor(tmax, 16));

            const float m_new = fmaxf(m_state, tmax);
            const float scale = __expf(m_state - m_new);

            float p[8];
            float psum = 0.0f;
#pragma unroll
            for (int j = 0; j < 8; ++j) {
                p[j] = __expf(c[j] - m_new);
                psum += p[j];
            }
            psum += __shfl_xor(psum, 16);
            l_state = l_state * scale + psum;
            m_state = m_new;

            // partial[h] = sum_j p[j] * V[t0 + j + 8*half][h]  (LDS broadcast)
            float part[HD];
#pragma unroll
            for (int hh = 0; hh < HD; ++hh) part[hh] = 0.0f;
#pragma unroll
            for (int j = 0; j < 8; ++j) {
                const float4* vrow =
                    (const float4*)&Vs[cur][(t0 + j + 8 * half) * HD];
                const float4 v0 = vrow[0], v1 = vrow[1];
                part[0] = fmaf(p[j], v0.x, part[0]);
                part[1] = fmaf(p[j], v0.y, part[1]);
                part[2] = fmaf(p[j], v0.z, part[2]);
                part[3] = fmaf(p[j], v0.w, part[3]);
                part[4] = fmaf(p[j], v1.x, part[4]);
                part[5] = fmaf(p[j], v1.y, part[5]);
                part[6] = fmaf(p[j], v1.z, part[6]);
                part[7] = fmaf(p[j], v1.w, part[7]);
            }
#pragma unroll
            for (int hh = 0; hh < HD; ++hh)
                acc[hh] = acc[hh] * scale + part[hh] + __shfl_xor(part[hh], 16);
        }

        // My async fills for the next buffer are done; barrier makes them
        // visible to all waves and confirms everyone finished this buffer.
        wait_async_lds();
        __syncthreads();
        cur ^= 1;
    }

    if (half == 0) {
        const float invl = 1.0f / l_state;
        float* orow = attn + ((size_t)b * SEQ + s0 + n) * HD;
        float4 o0, o1;
        o0.x = acc[0] * invl; o0.y = acc[1] * invl;
        o0.z = acc[2] * invl; o0.w = acc[3] * invl;
        o1.x = acc[4] * invl; o1.y = acc[5] * invl;
        o1.z = acc[6] * invl; o1.w = acc[7] * invl;
        ((float4*)orow)[0] = o0;
        ((float4*)orow)[1] = o1;
    }
}

// ---------------------------------------------------------------------------
// Kernel 3: out[r,:] = attn[r,0:8] @ Wo[8,1024] + bo. Bound by 64MB output
// write. Wo held in registers (8 x float4 per thread), attn row is a uniform
// load (scalarizable). 16 rows per block.
// ---------------------------------------------------------------------------
__global__ __launch_bounds__(256)
void out_proj(const float* __restrict__ attn, const float* __restrict__ Wo,
              const float* __restrict__ bo, float* __restrict__ out)
{
    const int r0 = blockIdx.x * 16;
    const int c4 = threadIdx.x;                    // float4 column 0..255
    const float4* Wo4 = (const float4*)Wo;

    float4 w[8];
#pragma unroll
    for (int h = 0; h < 8; ++h) w[h] = Wo4[h * (DMODEL / 4) + c4];
    const float4 ob = ((const float4*)bo)[c4];

    for (int r = 0; r < 16; ++r) {
        const float* ar = attn + (size_t)(r0 + r) * HD;
        float a[8];
#pragma unroll
        for (int h = 0; h < 8; ++h) a[h] = ar[h];
        float4 o = ob;
#pragma unroll
        for (int h = 0; h < 8; ++h) {
            o.x = fmaf(a[h], w[h].x, o.x);
            o.y = fmaf(a[h], w[h].y, o.y);
            o.z = fmaf(a[h], w[h].z, o.z);
            o.w = fmaf(a[h], w[h].w, o.w);
        }
        ((float4*)out)[(size_t)(r0 + r) * (DMODEL / 4) + c4] = o;
    }
}

// ---------------------------------------------------------------------------
extern "C" void kernel_launch(void* const* d_in, const int* in_sizes, int n_in,
                              void* d_out, int out_size, void* d_ws, size_t ws_size,
                              hipStream_t stream)
{
    const float* x       = (const float*)d_in[0];
    const float* context = (const float*)d_in[1];
    const float* Wq      = (const float*)d_in[2];
    const float* bq      = (const float*)d_in[3];
    const float* Wk      = (const float*)d_in[4];
    const float* bk      = (const float*)d_in[5];
    const float* Wv      = (const float*)d_in[6];
    const float* bv      = (const float*)d_in[7];
    const float* Wo      = (const float*)d_in[8];
    const float* bo      = (const float*)d_in[9];
    float* out = (float*)d_out;

    float* q    = (float*)d_ws;                // 16384*8 f32 = 512KB
    float* kbuf = q    + (size_t)ROWS * HD;
    float* vbuf = kbuf + (size_t)ROWS * HD;
    float* attn = vbuf + (size_t)ROWS * HD;    // total 2MB scratch

    // q from x (single output)
    proj_qkv<<<ROWS / 64, 128, 0, stream>>>(x, Wq, bq, q,
                                            nullptr, nullptr, nullptr, 0);
    // k and v fused from one read of context
    proj_qkv<<<ROWS / 64, 128, 0, stream>>>(context, Wk, bk, kbuf,
                                            Wv, bv, vbuf, 1);
    // flash attention (WMMA scores, async double-buffered K/V staging)
    attn_kernel<<<dim3(SEQ / 64, BATCH), 128, 0, stream>>>(q, kbuf, vbuf, attn);
    // output projection
    out_proj<<<ROWS / 16, 256, 0, stream>>>(attn, Wo, bo, out);
}